// GINConv_30288109371808
// MI455X (gfx1250) — compile-verified
//
#include <hip/hip_runtime.h>

typedef __attribute__((ext_vector_type(2))) float v2f;
typedef __attribute__((ext_vector_type(8))) float v8f;

#define D 256
#define LDS_STRIDE 260  // 256 + 4 floats pad: row stride advances banks by 4 -> conflict-free A reads

// ---------------------------------------------------------------------------
// Phase 1: scatter  agg[dst] += val * x[src]
// One lane per (edge, 4-dim chunk): 64 lanes/edge, float4 gather (coalesced),
// 4 x global_atomic_add_f32 into L2-resident agg.
// ---------------------------------------------------------------------------
__global__ __launch_bounds__(256) void gin_scatter(
    const float* __restrict__ x, const float* __restrict__ edge_val,
    const int* __restrict__ edge_src, const int* __restrict__ edge_dst,
    float* __restrict__ agg, int E) {
  long long tid = (long long)blockIdx.x * 256 + threadIdx.x;
  long long total = (long long)E * 64;
  if (tid >= total) return;
  int e  = (int)(tid >> 6);
  int d4 = (int)(tid & 63) << 2;
  int src = edge_src[e];
  int dst = edge_dst[e];
  float val = edge_val[e];
  float4 v = *reinterpret_cast<const float4*>(x + (size_t)src * D + d4);
  float* a = agg + (size_t)dst * D + d4;
  __hip_atomic_fetch_add(a + 0, val * v.x, __ATOMIC_RELAXED, __HIP_MEMORY_SCOPE_AGENT);
  __hip_atomic_fetch_add(a + 1, val * v.y, __ATOMIC_RELAXED, __HIP_MEMORY_SCOPE_AGENT);
  __hip_atomic_fetch_add(a + 2, val * v.z, __ATOMIC_RELAXED, __HIP_MEMORY_SCOPE_AGENT);
  __hip_atomic_fetch_add(a + 3, val * v.w, __ATOMIC_RELAXED, __HIP_MEMORY_SCOPE_AGENT);
}

// ---------------------------------------------------------------------------
// Phase 2: fused MLP.  One block = 16 output rows.  16 waves; wave w owns
// output columns [16w, 16w+16).  GEMM1 -> ReLU -> LDS -> GEMM2, all via
// V_WMMA_F32_16X16X4_F32 (exact fp32, matches fp32 reference).
// ---------------------------------------------------------------------------
__global__ __launch_bounds__(512) void gin_mlp(
    const float* __restrict__ x, const float* __restrict__ agg,
    const float* __restrict__ W1, const float* __restrict__ b1,
    const float* __restrict__ W2, const float* __restrict__ b2,
    const float* __restrict__ eps, float* __restrict__ out) {
  __shared__ float sup_s[16 * LDS_STRIDE];
  __shared__ float h_s[16 * LDS_STRIDE];

  const int tid  = threadIdx.x;
  const int wave = tid >> 5;        // 0..15
  const int lane = tid & 31;
  const int row0 = blockIdx.x * 16; // M-tile base (N is a multiple of 16)

  const float scale = 1.0f + eps[0];

  // ---- stage support tile [16 x 256] into LDS: each thread 8 floats ----
  {
    const int r = tid >> 5;          // row 0..15
    const int c = (tid & 31) * 8;    // col 0..248
    const float* xr = x   + (size_t)(row0 + r) * D + c;
    const float* ar = agg + (size_t)(row0 + r) * D + c;
    float* s = sup_s + r * LDS_STRIDE + c;
#pragma unroll
    for (int i = 0; i < 8; i += 4) {
      float4 xv = *reinterpret_cast<const float4*>(xr + i);
      float4 av = *reinterpret_cast<const float4*>(ar + i);
      s[i + 0] = scale * xv.x + av.x;
      s[i + 1] = scale * xv.y + av.y;
      s[i + 2] = scale * xv.z + av.z;
      s[i + 3] = scale * xv.w + av.w;
    }
  }
  __syncthreads();

  // A-matrix 16x4 f32 layout: lane<16 -> K=0,1 ; lane>=16 -> K=2,3 (row = lane&15)
  const int m   = lane & 15;          // A row within tile / B column within tile
  const int kh  = (lane >> 4) * 2;    // K sub-offset {0,2}
  const int col = wave * 16 + m;      // global output column for B / epilogue
  const int rbase = (lane >> 4) * 8;  // C/D: VGPR r -> row rbase + r

  // ---- GEMM1: h = relu(support @ W1^T + b1) -> LDS ----
  {
    v8f c = {};
    const float* wrow = W1 + (size_t)col * D;  // W1[col][k], row-major
#pragma unroll 8
    for (int k = 0; k < D; k += 4) {
      v2f a = *reinterpret_cast<const v2f*>(&sup_s[m * LDS_STRIDE + k + kh]);
      v2f b = *reinterpret_cast<const v2f*>(wrow + k + kh);
      c = __builtin_amdgcn_wmma_f32_16x16x4_f32(false, a, false, b,
                                                (short)0, c, false, false);
    }
    const float bias = b1[col];
#pragma unroll
    for (int r = 0; r < 8; ++r) {
      float v = c[r] + bias;
      h_s[(rbase + r) * LDS_STRIDE + col] = v > 0.0f ? v : 0.0f;
    }
  }
  __syncthreads();

  // ---- GEMM2: out = h @ W2^T + b2 ----
  {
    v8f c = {};
    const float* wrow = W2 + (size_t)col * D;
#pragma unroll 8
    for (int k = 0; k < D; k += 4) {
      v2f a = *reinterpret_cast<const v2f*>(&h_s[m * LDS_STRIDE + k + kh]);
      v2f b = *reinterpret_cast<const v2f*>(wrow + k + kh);
      c = __builtin_amdgcn_wmma_f32_16x16x4_f32(false, a, false, b,
                                                (short)0, c, false, false);
    }
    const float bias = b2[col];
#pragma unroll
    for (int r = 0; r < 8; ++r) {
      out[(size_t)(row0 + rbase + r) * D + col] = c[r] + bias;
    }
  }
}

// ---------------------------------------------------------------------------
extern "C" void kernel_launch(void* const* d_in, const int* in_sizes, int n_in,
                              void* d_out, int out_size, void* d_ws, size_t ws_size,
                              hipStream_t stream) {
  const float* x        = (const float*)d_in[0];
  const float* edge_val = (const float*)d_in[1];
  const float* W1       = (const float*)d_in[2];
  const float* b1       = (const float*)d_in[3];
  const float* W2       = (const float*)d_in[4];
  const float* b2       = (const float*)d_in[5];
  const float* eps      = (const float*)d_in[6];
  const int* edge_src   = (const int*)d_in[7];
  const int* edge_dst   = (const int*)d_in[8];
  float* out = (float*)d_out;
  float* agg = (float*)d_ws;  // [N x 256] f32 scratch

  const int E = in_sizes[1];
  const int N = in_sizes[0] / D;  // 100000

  // zero the aggregation buffer (capture-safe memset node)
  hipMemsetAsync(agg, 0, (size_t)N * D * sizeof(float), stream);

  // scatter: E * 64 lanes
  long long total = (long long)E * 64;
  int blocks = (int)((total + 255) / 256);
  gin_scatter<<<blocks, 256, 0, stream>>>(x, edge_val, edge_src, edge_dst, agg, E);

  // fused MLP: one block per 16 rows (N is a multiple of 16)
  gin_mlp<<<N / 16, 512, 0, stream>>>(x, agg, W1, b1, W2, b2, eps, out);
}